// Lookahead_21792664060574
// MI455X (gfx1250) — compile-verified
//
#include <hip/hip_runtime.h>
#include <stdint.h>

#define S_LEN   2048
#define BATCH   32
#define NFEAT   1024
#define K_TAPS  21
#define TROWS   21            // tile rows (one rotation period)
#define CHUNK   294           // 14 * 21, per-block sequence chunk
#define NTILES  15            // ceil((CHUNK + K_TAPS-1) / TROWS)
#define FH      512           // features per block
#define THREADS 256

#if __has_builtin(__builtin_amdgcn_tensor_load_to_lds)
#define USE_TDM 1
#else
#define USE_TDM 0
#endif

typedef __attribute__((ext_vector_type(4))) unsigned int tdm_u32x4;
typedef __attribute__((ext_vector_type(8))) int          tdm_i32x8;
typedef __attribute__((ext_vector_type(4))) int          tdm_i32x4;

__global__ __launch_bounds__(THREADS) void lookahead_tdm_kernel(
    const float* __restrict__ x, const float* __restrict__ w,
    float* __restrict__ out)
{
    __shared__ float tile[2][TROWS][FH];

    const int tid   = threadIdx.x;
    const int fhalf = blockIdx.x;          // 0..1
    const int b     = blockIdx.y;          // 0..31
    const int s0    = blockIdx.z * CHUNK;  // chunk start (multiple of 21)
    const int send  = min(S_LEN, s0 + CHUNK);
    const int f0    = fhalf * FH + 2 * tid;

    // Per-feature taps into VGPRs (each thread owns features f0, f0+1).
    float wx[K_TAPS], wy[K_TAPS];
#pragma unroll
    for (int k = 0; k < K_TAPS; ++k) {
        wx[k] = w[(size_t)f0 * K_TAPS + k];
        wy[k] = w[(size_t)(f0 + 1) * K_TAPS + k];
    }

    // Rotating accumulators: slot j holds the in-flight output with s % 21 == j.
    float ax[K_TAPS], ay[K_TAPS];
#pragma unroll
    for (int j = 0; j < K_TAPS; ++j) { ax[j] = 0.f; ay[j] = 0.f; }

    const bool issuer = (tid < 32);  // wave 0 drives the TDM

    // Global byte address of this block's (b, f-half) tensor view:
    // 2D tensor, dim0 = FH contiguous floats, row stride = BATCH*NFEAT floats.
    const uint64_t gbase =
        (uint64_t)(uintptr_t)(x + (size_t)b * NFEAT + (size_t)fhalf * FH);

#if USE_TDM
    auto issue_tile = [&](int i) {
        int t0  = s0 + i * TROWS;
        int t0c = (t0 < S_LEN) ? t0 : (S_LEN - 1);          // clamp fully-OOB tiles
        int rows = S_LEN - t0c; if (rows > TROWS) rows = TROWS;
        uint64_t ga = gbase + (uint64_t)t0c * (uint64_t)(BATCH * NFEAT) * 4ull;
        // Low 32 bits of a generic pointer to LDS == workgroup LDS byte offset.
        uint32_t lds_off = (uint32_t)(uintptr_t)&tile[i & 1][0][0];

        tdm_u32x4 g0;
        g0[0] = 1u;                                   // count=1, user descriptor
        g0[1] = lds_off;                              // lds_addr (bytes)
        g0[2] = (uint32_t)ga;                         // global_addr[31:0]
        g0[3] = (uint32_t)(ga >> 32) | (2u << 30);    // global_addr[56:32] | type=2

        tdm_i32x8 g1;
        g1[0] = (2 << 16);                            // wg_mask=0, data_size=4B
        g1[1] = (int)((FH & 0xFFFF) << 16);           // tensor_dim0[15:0]
        g1[2] = (int)(((uint32_t)rows & 0xFFFF) << 16); // dim0 hi=0 | tensor_dim1[15:0]
        g1[3] = (int)(FH << 16);                      // tensor_dim1 hi=0 | tile_dim0=FH
        g1[4] = rows;                                 // tile_dim1=rows, tile_dim2=0
        g1[5] = BATCH * NFEAT;                        // tensor_dim0_stride[31:0]
        g1[6] = 0;
        g1[7] = 0;

        tdm_i32x4 z4 = {0, 0, 0, 0};
        tdm_i32x8 z8 = {0, 0, 0, 0, 0, 0, 0, 0};
        __builtin_amdgcn_tensor_load_to_lds(g0, g1, z4, z4, z8, 0);
    };

    if (issuer) { issue_tile(0); issue_tile(1); }
#endif

    for (int i = 0; i < NTILES; ++i) {
        const int tbase = s0 + i * TROWS;

#if USE_TDM
        if (issuer) {
            if (i + 1 < NTILES) __builtin_amdgcn_s_wait_tensorcnt(1);
            else                __builtin_amdgcn_s_wait_tensorcnt(0);
        }
        __syncthreads();
#else
        // Fallback staging: each thread copies its own float2 per row.
        {
            int t0c = (tbase < S_LEN) ? tbase : (S_LEN - 1);
            int rows = S_LEN - t0c; if (rows > TROWS) rows = TROWS;
            const float* gsrc = (const float*)(uintptr_t)gbase +
                                (size_t)t0c * (BATCH * NFEAT) + 2 * tid;
#pragma unroll
            for (int u = 0; u < TROWS; ++u) {
                int r = (u < rows) ? u : (rows - 1);
                *(float2*)&tile[i & 1][u][2 * tid] =
                    *(const float2*)(gsrc + (size_t)r * (BATCH * NFEAT));
            }
            __syncthreads();
        }
#endif

        const float2* rp = (const float2*)&tile[i & 1][0][2 * tid];

#pragma unroll
        for (int u = 0; u < TROWS; ++u) {
            const int t = tbase + u;
            float2 xv = rp[u * (FH / 2)];
            if (t >= S_LEN) { xv.x = 0.f; xv.y = 0.f; }  // zero right-padding
#pragma unroll
            for (int k = 0; k < K_TAPS; ++k) {
                const int j = (u - k + 2 * K_TAPS) % K_TAPS;  // compile-time
                ax[j] = __builtin_fmaf(xv.x, wx[k], ax[j]);
                ay[j] = __builtin_fmaf(xv.y, wy[k], ay[j]);
            }
            const int jd = (u + 1) % K_TAPS;      // slot completed at step t
            const int s  = t - (K_TAPS - 1);
            if (s >= s0 && s < send) {
                float2 o; o.x = ax[jd]; o.y = ay[jd];
                *(float2*)(out + (size_t)s * (BATCH * NFEAT) +
                           (size_t)b * NFEAT + f0) = o;
            }
            ax[jd] = 0.f; ay[jd] = 0.f;           // recycle for s = t + 1
        }

        __syncthreads();
#if USE_TDM
        if (issuer && (i + 2 < NTILES)) issue_tile(i + 2);
#endif
    }
}

extern "C" void kernel_launch(void* const* d_in, const int* in_sizes, int n_in,
                              void* d_out, int out_size, void* d_ws, size_t ws_size,
                              hipStream_t stream) {
    const float* x = (const float*)d_in[0];
    const float* w = (const float*)d_in[1];
    float* out = (float*)d_out;
    dim3 grid(NFEAT / FH, BATCH, (S_LEN + CHUNK - 1) / CHUNK);  // 2 x 32 x 7
    lookahead_tdm_kernel<<<grid, THREADS, 0, stream>>>(x, w, out);
}